// Multi_attentive_feature_fusion_44985487458840
// MI455X (gfx1250) — compile-verified
//
#include <hip/hip_runtime.h>
#include <cstdint>

// ---------------------------------------------------------------------------
// Fused multi-attentive feature fusion, MI455X (gfx1250).
//
// Bandwidth-bound: 384 MB feature traffic @ 23.3 TB/s => ~17us floor.
// One streaming pass:
//   - TDM tensor_load_to_lds stages [32 rows x 128 ch] fp32 tiles per modality
//     (double buffered, wave0 issues, s_wait_tensorcnt + barriers sync).
//     Descriptor padding (128 DW interval, 4 DW amount) => LDS row stride of
//     132 floats => conflict-free ds_load_b64 for the WMMA A operand.
//   - v_wmma_f32_16x16x4_f32 accumulates all 15 per-row dot products in one
//     16x16 D accumulator: modality m owns N-columns [5m, 5m+5). B operand is
//     an LDS weight table Wt[128][16] rebuilt per chunk (zero elsewhere).
//   - Deterministic cross-wave serial LDS reduction + per-row softmax epilogue.
// ---------------------------------------------------------------------------

typedef float        v2f   __attribute__((ext_vector_type(2)));
typedef float        v8f   __attribute__((ext_vector_type(8)));
typedef unsigned int u32x4 __attribute__((ext_vector_type(4)));
typedef int          i32x8 __attribute__((ext_vector_type(8)));
typedef int          i32x4 __attribute__((ext_vector_type(4)));

#define C_IN     2048
#define ROWS_BLK 32
#define CHUNK_CH 128          // channels per TDM tile
#define TILE_STRIDE 132       // 128 + 4 pad floats per row (from TDM pad)
#define N_CHUNK  48           // 3 modalities * (2048/128)

// Issue one TDM 2D tile load: [32 rows x 128 ch] fp32, global row stride 2048,
// LDS destination with +4-DWORD pad after every 128 DWORDs stored.
__device__ __forceinline__ void tdm_load_tile(const float* gsrc, unsigned ldsOff) {
    unsigned long long ga = (unsigned long long)(uintptr_t)gsrc;
    u32x4 g0;
    g0[0] = 1u;                                          // count=1, user desc
    g0[1] = ldsOff;                                      // lds_addr (bytes)
    g0[2] = (unsigned)(ga & 0xFFFFFFFFull);              // global_addr[31:0]
    g0[3] = (unsigned)((ga >> 32) & 0x01FFFFFFull)       // global_addr[56:32]
          | (2u << 30);                                  // type=2 ("image")
    i32x8 g1;
    g1[0] = (int)((2u << 16)      // data_size = 4 bytes
                | (1u << 20)      // pad_enable
                | (6u << 22)      // pad_interval: 128 DWORDs
                | (3u << 25));    // pad_amount: 4 DWORDs
    g1[1] = (int)(2048u << 16);   // tensor_dim0[15:0] = 2048 (bits 63:48)
    g1[2] = (int)(32u << 16);     // dim0 hi=0 | tensor_dim1[15:0]=32
    g1[3] = (int)(128u << 16);    // dim1 hi=0 | tile_dim0=128
    g1[4] = 32;                   // tile_dim1=32 | tile_dim2=0
    g1[5] = 2048;                 // tensor_dim0_stride[31:0]
    g1[6] = 0;                    // stride hi | tensor_dim1_stride lo (unused, 2D)
    g1[7] = 0;
    i32x4 gz4 = {0, 0, 0, 0};             // groups 2/3 unused (2D tensor)
    i32x8 gz8 = {0, 0, 0, 0, 0, 0, 0, 0}; // extra group (clang-23 6-arg form)
    __builtin_amdgcn_tensor_load_to_lds(g0, g1, gz4, gz4, gz8, 0);
}

__global__ __launch_bounds__(256)
void maff_fused_kernel(const float* __restrict__ s_f,
                       const float* __restrict__ g_f,
                       const float* __restrict__ v_f,
                       const float* __restrict__ W_aff,
                       const float* __restrict__ b_aff,
                       const float* __restrict__ W_cls,
                       const float* __restrict__ b_cls,
                       float* __restrict__ out) {
    __shared__ float tileA[2][ROWS_BLK * TILE_STRIDE];   // 2 x 16.5 KB feature tiles
    __shared__ float Wt[CHUNK_CH * 16];                  // 8 KB arranged weight table
    __shared__ float red8[8 * ROWS_BLK * 16];            // 16 KB per-wave partials

    const int tid  = threadIdx.x;
    const int lane = tid & 31;
    const int wid  = tid >> 5;            // wave id 0..7 == 16-channel window
    const int mrow = lane & 15;           // WMMA M (A) and N (B/D) lane index
    const int half = lane >> 4;           // K-half select (K+0/1 vs K+2/3)
    const int row0 = blockIdx.x * ROWS_BLK;

    const float* srcs[3] = {s_f, g_f, v_f};

    // Prefetch chunk 0 (modality 0, channels 0..127).
    if (wid == 0) {
        tdm_load_tile(srcs[0] + (size_t)row0 * C_IN,
                      (unsigned)(uintptr_t)&tileA[0][0]);
    }

    v8f acc0 = {0.f, 0.f, 0.f, 0.f, 0.f, 0.f, 0.f, 0.f};  // rows  0..15
    v8f acc1 = {0.f, 0.f, 0.f, 0.f, 0.f, 0.f, 0.f, 0.f};  // rows 16..31

    for (int q = 0; q < N_CHUNK; ++q) {
        const int m  = q >> 4;            // modality
        const int cb = q & 15;            // channel block
        const int c0 = cb * CHUNK_CH;

        // Wave 0: issue next tile, then wait until chunk q's tile has landed.
        if (wid == 0) {
            if (q + 1 < N_CHUNK) {
                const int mn  = (q + 1) >> 4;
                const int cbn = (q + 1) & 15;
                tdm_load_tile(srcs[mn] + (size_t)row0 * C_IN + cbn * CHUNK_CH,
                              (unsigned)(uintptr_t)&tileA[(q + 1) & 1][0]);
                __builtin_amdgcn_s_wait_tensorcnt(1);   // chunk q done, q+1 in flight
            } else {
                __builtin_amdgcn_s_wait_tensorcnt(0);
            }
        }

        // Build B weight table for this chunk: Wt[ch][n]; modality m owns
        // columns [5m,5m+3) = W_aff cols, [5m+3,5m+5) = W_cls cols, else 0.
        {
            const int chl     = tid >> 1;                 // 0..127
            const int colBase = (tid & 1) * 8;
            const int wrow    = m * C_IN + c0 + chl;      // row in [3C, x] weights
            #pragma unroll
            for (int i = 0; i < 8; ++i) {
                const int col = colBase + i;
                const int rel = col - m * 5;
                float w = 0.f;
                if (rel >= 0 && rel < 3)       w = W_aff[wrow * 3 + rel];
                else if (rel == 3 || rel == 4) w = W_cls[wrow * 2 + (rel - 3)];
                Wt[chl * 16 + col] = w;
            }
        }
        __syncthreads();   // tile q ready (wave0 waited) + Wt published

        // Compute: wave `wid` covers channels [wid*16, wid*16+16) of the chunk.
        {
            const float* tp  = &tileA[q & 1][0];
            const int    wch = wid * 16;
            #pragma unroll
            for (int k0 = 0; k0 < 16; k0 += 4) {
                const int ch = wch + k0 + 2 * half;       // this lane's K channel
                v2f b;
                b.x = Wt[ch * 16 + mrow];                 // B[K  ][n]
                b.y = Wt[(ch + 1) * 16 + mrow];           // B[K+1][n]
                v2f a0 = *(const v2f*)&tp[mrow * TILE_STRIDE + ch];
                v2f a1 = *(const v2f*)&tp[(16 + mrow) * TILE_STRIDE + ch];
                acc0 = __builtin_amdgcn_wmma_f32_16x16x4_f32(
                           false, a0, false, b, (short)0, acc0, false, false);
                acc1 = __builtin_amdgcn_wmma_f32_16x16x4_f32(
                           false, a1, false, b, (short)0, acc1, false, false);
            }
        }
        __syncthreads();   // all waves done with tile[q&1] and Wt
    }

    // Deterministic cross-wave reduction: each wave dumps its D to a private
    // slab; D layout: VGPR r, lane -> M = r + 8*half (+16 for acc1), N = mrow.
    {
        float* slab = &red8[wid * (ROWS_BLK * 16)];
        #pragma unroll
        for (int r = 0; r < 8; ++r) {
            const int M = r + 8 * half;
            slab[M * 16 + mrow]        = acc0[r];
            slab[(16 + M) * 16 + mrow] = acc1[r];
        }
    }
    __syncthreads();

    // Epilogue: one thread per row -> softmax(att over 3) -> gate -> softmax(2).
    if (tid < ROWS_BLK) {
        const int row = tid;
        float rd[15];
        #pragma unroll
        for (int i = 0; i < 15; ++i) {
            float acc = 0.f;
            #pragma unroll
            for (int w = 0; w < 8; ++w)
                acc += red8[w * (ROWS_BLK * 16) + row * 16 + i];
            rd[i] = acc;
        }
        // Attention logits: sum the three modality slices of W_aff.
        float L0 = rd[0] + rd[5] + rd[10] + b_aff[0];
        float L1 = rd[1] + rd[6] + rd[11] + b_aff[1];
        float L2 = rd[2] + rd[7] + rd[12] + b_aff[2];
        float mx = fmaxf(L0, fmaxf(L1, L2));
        float e0 = expf(L0 - mx), e1 = expf(L1 - mx), e2 = expf(L2 - mx);
        float inv = 1.f / (e0 + e1 + e2);
        float a0 = e0 * inv, a1 = e1 * inv, a2 = e2 * inv;
        // Classifier logits with per-modality gating (linearity of the dot).
        float y0 = a0 * rd[3] + a1 * rd[8] + a2 * rd[13] + b_cls[0];
        float y1 = a0 * rd[4] + a1 * rd[9] + a2 * rd[14] + b_cls[1];
        float my = fmaxf(y0, y1);
        float f0 = expf(y0 - my), f1 = expf(y1 - my);
        float is = 1.f / (f0 + f1);
        float2 o; o.x = f0 * is; o.y = f1 * is;
        *(float2*)&out[(size_t)(row0 + row) * 2] = o;
    }
}

extern "C" void kernel_launch(void* const* d_in, const int* in_sizes, int n_in,
                              void* d_out, int out_size, void* d_ws, size_t ws_size,
                              hipStream_t stream) {
    const float* s_f   = (const float*)d_in[0];
    const float* g_f   = (const float*)d_in[1];
    const float* v_f   = (const float*)d_in[2];
    const float* W_aff = (const float*)d_in[3];
    const float* b_aff = (const float*)d_in[4];
    const float* W_cls = (const float*)d_in[5];
    const float* b_cls = (const float*)d_in[6];
    float* out = (float*)d_out;

    const int rows   = in_sizes[0] / C_IN;       // 16384
    const int blocks = rows / ROWS_BLK;          // 512

    maff_fused_kernel<<<dim3(blocks), dim3(256), 0, stream>>>(
        s_f, g_f, v_f, W_aff, b_aff, W_cls, b_cls, out);
}